// LogSigRNN_15375982919959
// MI455X (gfx1250) — compile-verified
//
#include <hip/hip_runtime.h>
#include <hip/hip_bf16.h>

// ---------------------------------------------------------------------------
// LogSigRNN for MI455X (gfx1250, wave32).
//   * schedule via exact integer math on device
//   * logsig via telescoped antisymmetric sum, x[b] staged in LDS
//   * gate-independent input projection hoisted: G = U @ W_r0[256:] + b_r0
//   * all GEMMs on V_WMMA_F32_16X16X4_F32; weights pre-packed into B-fragment
//     order so the inner loop is 1 coalesced b64 A-load + RN b64 B-loads +
//     RN WMMAs (RN = 4 or 2 N-tiles per wave; A fragment reused across them)
// ---------------------------------------------------------------------------

typedef __attribute__((ext_vector_type(2))) float v2f;
typedef __attribute__((ext_vector_type(8))) float v8f;

#define ACT_NONE 0
#define ACT_PRELU 1
#define ACT_TANH 2

#define Bsz 256
#define NT 256
#define Dd 32
#define HID 256
#define NLAG 64
#define UCOLS 560   // st(32) + x[ind_max](32) + Ltri(496)

// ---------------------------------------------------------------------------
// schedule: ind_max[i]=floor(85i/21); k=floor(85i/336); ind_low=16k;
// gate[i] = (i==0) || k(i)>k(i-1) || (i==63)
// ---------------------------------------------------------------------------
__global__ void sched_kernel(int* __restrict__ ind_low, int* __restrict__ ind_max,
                             int* __restrict__ gate) {
    int i = threadIdx.x;
    if (i >= NLAG) return;
    int k  = (85 * i) / 336;
    int km = (i > 0) ? (85 * (i - 1)) / 336 : -1;
    ind_max[i] = (85 * i) / 21;
    ind_low[i] = 16 * k;
    gate[i]    = (i == 0 || k > km || i == NLAG - 1) ? 1 : 0;
}

// ---------------------------------------------------------------------------
// logsig: per batch row, U[b][l][0:560] = [x[lo], x[hi], Ltri(496)]
// L[i][j] = 0.5 * sum_{t=lo}^{hi-1} (x_t[i]*x_{t+1}[j] - x_t[j]*x_{t+1}[i])
// ---------------------------------------------------------------------------
__global__ void logsig_kernel(const float* __restrict__ x,
                              const int* __restrict__ ind_low,
                              const int* __restrict__ ind_max,
                              float* __restrict__ U) {
    __shared__ float xs[NT * Dd];   // 32 KB of the 320 KB WGP LDS
    const int b = blockIdx.x;
    for (int i = threadIdx.x; i < NT * Dd; i += blockDim.x)
        xs[i] = x[(size_t)b * NT * Dd + i];
    __syncthreads();

    for (int l = 0; l < NLAG; ++l) {
        const int lo = ind_low[l], hi = ind_max[l];
        float* u = U + ((size_t)b * NLAG + l) * UCOLS;
        if (threadIdx.x < Dd)          u[threadIdx.x] = xs[lo * Dd + threadIdx.x];
        else if (threadIdx.x < 2 * Dd) u[threadIdx.x] = xs[hi * Dd + threadIdx.x - Dd];
        for (int p = threadIdx.x; p < 496; p += blockDim.x) {
            int rem = p, i = 0;
            while (rem >= 31 - i) { rem -= 31 - i; ++i; }
            const int j = i + 1 + rem;
            float s = 0.f;
            for (int t = lo; t < hi; ++t) {
                s += xs[t * Dd + i] * xs[(t + 1) * Dd + j]
                   - xs[t * Dd + j] * xs[(t + 1) * Dd + i];
            }
            u[2 * Dd + p] = 0.5f * s;
        }
    }
}

// ---------------------------------------------------------------------------
// Pack W (K,N row-major) into B-fragment order:
//   Wp[ ((nt*(K/4)+kq)*32 + lane)*2 + v ] = W[(4*kq + 2*(lane>>4) + v)*N + nt*16 + (lane&15)]
// so each lane's per-k-step B fragment is one contiguous b64, and a wave's
// 32 lanes cover one 256 B packed record (fully coalesced).
// ---------------------------------------------------------------------------
__global__ void pack_w(const float* __restrict__ W, float* __restrict__ Wp,
                       int K, int N) {
    int idx = blockIdx.x * blockDim.x + threadIdx.x;
    if (idx >= K * N) return;
    const int v    = idx & 1;
    const int lane = (idx >> 1) & 31;
    const int rest = idx >> 6;
    const int K4   = K >> 2;
    const int kq   = rest % K4;
    const int nt   = rest / K4;
    const int krow = 4 * kq + 2 * (lane >> 4) + v;
    const int ncol = nt * 16 + (lane & 15);
    Wp[idx] = W[(size_t)krow * N + ncol];
}

// ---------------------------------------------------------------------------
// Packed fp32 WMMA GEMM: C = act(A@W + bias + cinit) [+ residual]
// One wave computes a 16 x (16*RN) strip of C; A fragment loaded once per
// k-step and reused for RN WMMAs.
// Fragment layouts per CDNA5 ISA 7.12.2 (wave32):
//   A 16x4 : lane m=l&15, half=l>>4; vgpr v holds A[m][2*half+v]
//   B 4x16 : lane n=l&15, half=l>>4; vgpr v holds B[2*half+v][n]
//   C 16x16: lane n=l&15; vgpr v holds C[v + 8*(l>>4)][n]
// ---------------------------------------------------------------------------
template <int RN>
__global__ void gemm16p_wmma(const float* __restrict__ A, int lda,
                             const float* __restrict__ Wp, int K,
                             const float* __restrict__ bias,
                             const float* __restrict__ cinit, int ldci,
                             const float* __restrict__ resid, int ldres,
                             float* __restrict__ C, int ldc,
                             int M, int N,
                             int act, const float* __restrict__ alphap) {
    const int tilesN = N >> 4;
    const int groups = tilesN / RN;
    const int wave   = blockIdx.x * (blockDim.x >> 5) + (threadIdx.x >> 5);
    const int tileM  = wave / groups;
    const int ng     = wave % groups;
    if (tileM >= (M >> 4)) return;        // wave-uniform: EXEC stays all-ones

    const int lane = threadIdx.x & 31;
    const int half = lane >> 4;           // which K pair this lane holds
    const int mr   = lane & 15;           // A row / B-C column within tile
    const int K4   = K >> 2;

    const float* Arow = A + (size_t)(tileM * 16 + mr) * lda + 2 * half;
    const float* Bp   = Wp + (size_t)(ng * RN) * K4 * 64 + lane * 2;

    v8f acc[RN];
    #pragma unroll
    for (int r = 0; r < RN; ++r) acc[r] = (v8f){};

    for (int kq = 0; kq < K4; ++kq) {
        __builtin_prefetch(Arow + 4 * kq + 64, 0, 0);     // global_prefetch_b8
        const v2f a = *(const v2f*)(Arow + 4 * kq);
        #pragma unroll
        for (int r = 0; r < RN; ++r) {
            const v2f b = *(const v2f*)(Bp + ((size_t)r * K4 + kq) * 64);
            acc[r] = __builtin_amdgcn_wmma_f32_16x16x4_f32(
                /*neg_a=*/false, a, /*neg_b=*/false, b,
                /*c_mod=*/(short)0, acc[r], /*reuse_a=*/false, /*reuse_b=*/false);
        }
    }

    const float alpha = alphap ? alphap[0] : 0.0f;
    #pragma unroll
    for (int r = 0; r < RN; ++r) {
        const int n = (ng * RN + r) * 16 + mr;
        #pragma unroll
        for (int v = 0; v < 8; ++v) {
            const int m = tileM * 16 + v + 8 * half;
            float val = acc[r][v];
            if (cinit) val += cinit[(size_t)m * ldci + n];
            if (bias)  val += bias[n];
            if (act == ACT_PRELU)     val = (val >= 0.f) ? val : alpha * val;
            else if (act == ACT_TANH) val = tanhf(val);
            if (resid) val += resid[(size_t)m * ldres + n];
            C[(size_t)m * ldc + n] = val;
        }
    }
}

// ---------------------------------------------------------------------------
// carry update: h_last = gate[l] ? hcur : h_last
// ---------------------------------------------------------------------------
__global__ void gate_update(float* __restrict__ hlast, const float* __restrict__ hcur,
                            const int* __restrict__ gate, int l, int n) {
    if (!gate[l]) return;                 // uniform across grid
    int i = blockIdx.x * blockDim.x + threadIdx.x;
    if (i < n) hlast[i] = hcur[i];
}

// ---------------------------------------------------------------------------
static inline void launch_pack(hipStream_t s, const float* W, float* Wp, int K, int N) {
    pack_w<<<(K * N + 255) / 256, 256, 0, s>>>(W, Wp, K, N);
}

static inline void launch_gemm(hipStream_t s,
                               const float* A, int lda, const float* Wp, int K,
                               const float* bias,
                               const float* cinit, int ldci,
                               const float* resid, int ldres,
                               float* C, int ldc, int M, int N,
                               int act, const float* alphap) {
    const int tilesN = N / 16;
    if ((tilesN & 3) == 0) {
        const int waves = (M / 16) * (tilesN / 4);
        gemm16p_wmma<4><<<(waves + 7) / 8, 256, 0, s>>>(A, lda, Wp, K, bias,
            cinit, ldci, resid, ldres, C, ldc, M, N, act, alphap);
    } else {
        const int waves = (M / 16) * (tilesN / 2);
        gemm16p_wmma<2><<<(waves + 7) / 8, 256, 0, s>>>(A, lda, Wp, K, bias,
            cinit, ldci, resid, ldres, C, ldc, M, N, act, alphap);
    }
}

extern "C" void kernel_launch(void* const* d_in, const int* in_sizes, int n_in,
                              void* d_out, int out_size, void* d_ws, size_t ws_size,
                              hipStream_t stream) {
    const float* x    = (const float*)d_in[0];
    const float* W_i0 = (const float*)d_in[1];
    const float* b_i0 = (const float*)d_in[2];
    const float* a_i0 = (const float*)d_in[3];
    const float* W_i1 = (const float*)d_in[4];
    const float* b_i1 = (const float*)d_in[5];
    const float* a_i1 = (const float*)d_in[6];
    const float* W_i2 = (const float*)d_in[7];
    const float* b_i2 = (const float*)d_in[8];
    const float* W_r0 = (const float*)d_in[9];   // (816, 256)
    const float* b_r0 = (const float*)d_in[10];
    const float* a_r0 = (const float*)d_in[11];
    const float* W_r1 = (const float*)d_in[12];
    const float* b_r1 = (const float*)d_in[13];
    const float* a_r1 = (const float*)d_in[14];
    const float* W_r2 = (const float*)d_in[15];
    const float* b_r2 = (const float*)d_in[16];
    const float* W_out = (const float*)d_in[17]; // (256, 32)
    float* out = (float*)d_out;                  // (B, 64, 32)

    // workspace carve-out (~55 MB)
    char*  ws      = (char*)d_ws;
    int*   ind_low = (int*)ws;
    int*   ind_max = ind_low + 64;
    int*   gate    = ind_max + 64;
    float* U    = (float*)(ws + 1024);                     // (B*64, 560)
    float* G    = U   + (size_t)Bsz * NLAG * UCOLS;        // (B*64, 256)
    float* h1   = G   + (size_t)Bsz * NLAG * HID;          // (B, 256)
    float* h2   = h1  + (size_t)Bsz * HID;
    float* hl   = h2  + (size_t)Bsz * HID;                 // carry
    float* z1   = hl  + (size_t)Bsz * HID;
    float* z2   = z1  + (size_t)Bsz * HID;
    float* hc   = z2  + (size_t)Bsz * HID;
    float* Pi0  = hc  + (size_t)Bsz * HID;                 // packed weights:
    float* Pi1  = Pi0 + (size_t)Dd * HID;                  //  W_i0 (32,256)
    float* Pi2  = Pi1 + (size_t)HID * HID;                 //  W_i1 (256,256)
    float* Pr0h = Pi2 + (size_t)HID * HID;                 //  W_i2 (256,256)
    float* Pr0u = Pr0h + (size_t)HID * HID;                //  W_r0[:256]
    float* Pr1  = Pr0u + (size_t)UCOLS * HID;              //  W_r0[256:]
    float* Pr2  = Pr1 + (size_t)HID * HID;
    float* Pout = Pr2 + (size_t)HID * HID;                 //  W_out (256,32)

    // 0) pack all weights into B-fragment order (amortized over 64 steps)
    launch_pack(stream, W_i0, Pi0, Dd, HID);
    launch_pack(stream, W_i1, Pi1, HID, HID);
    launch_pack(stream, W_i2, Pi2, HID, HID);
    launch_pack(stream, W_r0, Pr0h, HID, HID);             // rows 0..255
    launch_pack(stream, W_r0 + (size_t)HID * HID, Pr0u, UCOLS, HID); // rows 256..815
    launch_pack(stream, W_r1, Pr1, HID, HID);
    launch_pack(stream, W_r2, Pr2, HID, HID);
    launch_pack(stream, W_out, Pout, HID, Dd);

    // 1) schedule
    sched_kernel<<<1, 64, 0, stream>>>(ind_low, ind_max, gate);

    // 2) logsig features U = [starts | x[ind_max] | Ltri]
    logsig_kernel<<<Bsz, 256, 0, stream>>>(x, ind_low, ind_max, U);

    // 3) G = U @ W_r0[256:816] + b_r0   (16384 x 560 x 256)
    launch_gemm(stream, U, UCOLS, Pr0u, UCOLS, b_r0, nullptr, 0, nullptr, 0,
                G, HID, Bsz * NLAG, HID, ACT_NONE, nullptr);

    // 4) init MLP
    launch_gemm(stream, x, NT * Dd, Pi0, Dd, b_i0, nullptr, 0, nullptr, 0,
                h1, HID, Bsz, HID, ACT_PRELU, a_i0);
    launch_gemm(stream, h1, HID, Pi1, HID, b_i1, nullptr, 0, h1, HID,
                h2, HID, Bsz, HID, ACT_PRELU, a_i1);
    launch_gemm(stream, h2, HID, Pi2, HID, b_i2, nullptr, 0, nullptr, 0,
                hl, HID, Bsz, HID, ACT_TANH, nullptr);

    // 5) recurrent scan (sequential only through hl)
    for (int l = 0; l < NLAG; ++l) {
        launch_gemm(stream, hl, HID, Pr0h, HID, nullptr,
                    G + (size_t)l * HID, NLAG * HID, nullptr, 0,
                    z1, HID, Bsz, HID, ACT_PRELU, a_r0);
        launch_gemm(stream, z1, HID, Pr1, HID, b_r1, nullptr, 0, nullptr, 0,
                    z2, HID, Bsz, HID, ACT_PRELU, a_r1);
        launch_gemm(stream, z2, HID, Pr2, HID, b_r2, nullptr, 0, nullptr, 0,
                    hc, HID, Bsz, HID, ACT_TANH, nullptr);
        launch_gemm(stream, hc, HID, Pout, HID, nullptr, nullptr, 0, nullptr, 0,
                    out + (size_t)l * Dd, NLAG * Dd, Bsz, Dd, ACT_NONE, nullptr);
        gate_update<<<(Bsz * HID + 255) / 256, 256, 0, stream>>>(hl, hc, gate, l, Bsz * HID);
    }
}